// TriFreqConv_25563645346255
// MI455X (gfx1250) — compile-verified
//
#include <hip/hip_runtime.h>
#include <cstdint>

#define HH 256
#define WW 256
#define TS 32
#define HALO 4
#define LT 40            // TS + 2*HALO
#define NCH 96

// ---- CDNA5 async global->LDS staging (ASYNCcnt path) ----
__device__ __forceinline__ void async_ld_b128(uint32_t lds_off, const float* g) {
    asm volatile("global_load_async_to_lds_b128 %0, %1, off"
                 :: "v"(lds_off), "v"((unsigned long long)(uintptr_t)g)
                 : "memory");
}
__device__ __forceinline__ void wait_async0() {
    asm volatile("s_wait_asynccnt 0x0" ::: "memory");
}

__device__ __forceinline__ float pool_cnt(int gy, int gx, int p) {
    int y0 = gy - p; if (y0 < 0) y0 = 0;
    int y1 = gy + p; if (y1 > HH - 1) y1 = HH - 1;
    int x0 = gx - p; if (x0 < 0) x0 = 0;
    int x1 = gx + p; if (x1 > WW - 1) x1 = WW - 1;
    return (float)((y1 - y0 + 1) * (x1 - x0 + 1));
}

// BN + SiLU + streaming (non-temporal) store: output is write-once, keep L2 for input.
__device__ __forceinline__ void bn_silu_store(float acc, float inv, float bb, float* p) {
    float y = acc * inv + bb;
    __builtin_nontemporal_store(y / (1.f + __expf(-y)), p);
}

__global__ __launch_bounds__(256)
void trifreq_kernel(const float* __restrict__ x,
                    const float* __restrict__ w_lo,
                    const float* __restrict__ w_mf,
                    const float* __restrict__ w_hf,
                    const float* __restrict__ bn_g,
                    const float* __restrict__ bn_b,
                    const float* __restrict__ bn_m,
                    const float* __restrict__ bn_v,
                    float* __restrict__ out)
{
    __shared__ __align__(16) float s_in[LT * LT];   // 40x40 input tile (stride 40, 16B rows)
    __shared__ float s_r3[40 * 36];                 // row sums width-3 (rows 0..39, cols 0..33)
    __shared__ float s_r7[40 * 36];                 // row sums width-7
    __shared__ float s_t [34 * 36];                 // band/high-pass intermediate (34x34)

    const int tid = threadIdx.x;
    const int bx  = blockIdx.x, by = blockIdx.y;
    const int bc  = blockIdx.z;                 // b*96 + c
    const int c   = bc % NCH;
    const int b   = bc / NCH;
    const int branch = c >> 5;                  // 0=lo(5x5), 1=mf(bandpass), 2=hf(highpass)
    const int cl  = c & 31;
    const int tx0 = bx * TS, ty0 = by * TS;

    // ---------- stage 40x40 tile into LDS via async b128 (400 chunks) ----------
    const float* xc = x + (size_t)bc * (HH * WW);
    const uint32_t lds_base = (uint32_t)(uintptr_t)(&s_in[0]);
    for (int idx = tid; idx < 400; idx += 256) {
        int r  = idx / 10;
        int ch = idx - r * 10;
        int gy = ty0 - HALO + r;
        int gx = tx0 - HALO + 4 * ch;           // multiple of 4 -> chunk fully in or out
        uint32_t loff = lds_base + (uint32_t)((r * LT + 4 * ch) * 4);
        if ((unsigned)gy < HH && (unsigned)gx < WW) {
            async_ld_b128(loff, xc + gy * WW + gx);
        } else {
            *(float4*)&s_in[r * LT + 4 * ch] = make_float4(0.f, 0.f, 0.f, 0.f);
        }
    }
    wait_async0();
    __syncthreads();

    // ---------- per-block uniform BN constants (channel shuffle: oc = cl*3 + branch) ----------
    const int oc = cl * 3 + branch;
    const float inv = bn_g[oc] * rsqrtf(bn_v[oc] + 1e-5f);
    const float bb  = bn_b[oc] - bn_m[oc] * inv;
    float* op = out + ((size_t)b * NCH + oc) * (HH * WW) + (size_t)ty0 * WW + tx0;

    if (branch == 0) {
        // ================= 5x5 depthwise conv =================
        float w[25];
        #pragma unroll
        for (int i = 0; i < 25; ++i) w[i] = w_lo[cl * 25 + i];
        #pragma unroll
        for (int j = 0; j < 4; ++j) {
            int r   = (tid >> 5) + 8 * j;
            int col =  tid & 31;
            float acc = 0.f;
            #pragma unroll
            for (int dy = 0; dy < 5; ++dy)
                #pragma unroll
                for (int dx = 0; dx < 5; ++dx)
                    acc = fmaf(w[dy * 5 + dx], s_in[(r + dy + 2) * LT + (col + dx + 2)], acc);
            bn_silu_store(acc, inv, bb, op + r * WW + col);
        }
        return;
    }

    if (branch == 1) {
        // ================= band-pass: ap3 - ap7, then 3x3 conv =================
        // phase A: separable row sums over cols c' in [-1,32] (cc = c'+1 in [0,34))
        for (int idx = tid; idx < 40 * 34; idx += 256) {
            int rr = idx / 34, cc = idx - rr * 34;
            const float* row = &s_in[rr * LT];
            float a = row[cc],     b0 = row[cc + 1], c0 = row[cc + 2], d = row[cc + 3];
            float e = row[cc + 4], f  = row[cc + 5], g = row[cc + 6];
            s_r3[rr * 36 + cc] = c0 + d + e;
            s_r7[rr * 36 + cc] = ((a + b0) + (c0 + d)) + ((e + f) + g);
        }
        __syncthreads();
        // phase B: column sums -> bp on 34x34 region (r' in [-1,32], rp = r'+1)
        for (int idx = tid; idx < 34 * 34; idx += 256) {
            int rp = idx / 34, cc = idx - rp * 34;
            int gy = ty0 + rp - 1, gx = tx0 + cc - 1;
            float v = 0.f;
            if ((unsigned)gy < HH && (unsigned)gx < WW) {
                float s3 = s_r3[(rp + 2) * 36 + cc] + s_r3[(rp + 3) * 36 + cc]
                         + s_r3[(rp + 4) * 36 + cc];
                float s7 = 0.f;
                #pragma unroll
                for (int d = 0; d < 7; ++d) s7 += s_r7[(rp + d) * 36 + cc];
                v = s3 / pool_cnt(gy, gx, 1) - s7 / pool_cnt(gy, gx, 3);
            }
            s_t[rp * 36 + cc] = v;   // zero at out-of-image -> conv zero padding
        }
        __syncthreads();
    } else {
        // ================= high-pass: x - ap3, then 3x3 conv =================
        for (int idx = tid; idx < 34 * 34; idx += 256) {
            int rp = idx / 34, cc = idx - rp * 34;
            int gy = ty0 + rp - 1, gx = tx0 + cc - 1;
            float v = 0.f;
            if ((unsigned)gy < HH && (unsigned)gx < WW) {
                int ri = rp + 3, ci = cc + 3;        // tile index of (r',c')
                float s = 0.f;
                #pragma unroll
                for (int dy = -1; dy <= 1; ++dy)
                    #pragma unroll
                    for (int dx = -1; dx <= 1; ++dx)
                        s += s_in[(ri + dy) * LT + (ci + dx)];
                v = s_in[ri * LT + ci] - s / pool_cnt(gy, gx, 1);
            }
            s_t[rp * 36 + cc] = v;
        }
        __syncthreads();
    }

    // ---------- shared 3x3 depthwise conv epilogue (mf / hf) ----------
    const float* w9 = (branch == 1 ? w_mf : w_hf) + cl * 9;
    float w[9];
    #pragma unroll
    for (int i = 0; i < 9; ++i) w[i] = w9[i];
    #pragma unroll
    for (int j = 0; j < 4; ++j) {
        int r   = (tid >> 5) + 8 * j;
        int col =  tid & 31;
        float acc = 0.f;
        #pragma unroll
        for (int dy = 0; dy < 3; ++dy)
            #pragma unroll
            for (int dx = 0; dx < 3; ++dx)
                acc = fmaf(w[dy * 3 + dx], s_t[(r + dy) * 36 + (col + dx)], acc);
        bn_silu_store(acc, inv, bb, op + r * WW + col);
    }
}

extern "C" void kernel_launch(void* const* d_in, const int* in_sizes, int n_in,
                              void* d_out, int out_size, void* d_ws, size_t ws_size,
                              hipStream_t stream) {
    const float* x    = (const float*)d_in[0];
    const float* w_lo = (const float*)d_in[1];
    const float* w_mf = (const float*)d_in[2];
    const float* w_hf = (const float*)d_in[3];
    const float* g    = (const float*)d_in[4];
    const float* be   = (const float*)d_in[5];
    const float* mu   = (const float*)d_in[6];
    const float* va   = (const float*)d_in[7];
    float* out = (float*)d_out;

    dim3 grid(WW / TS, HH / TS, 8 * NCH);   // 8x8 tiles, 8 batches * 96 channels
    trifreq_kernel<<<grid, 256, 0, stream>>>(x, w_lo, w_mf, w_hf, g, be, mu, va, out);
}